// MainVoxelEncoder_35150012350922
// MI455X (gfx1250) — compile-verified
//
#include <hip/hip_runtime.h>
#include <hip/hip_bf16.h>

typedef float v2f __attribute__((ext_vector_type(2)));
typedef float v4f __attribute__((ext_vector_type(4)));
typedef float v8f __attribute__((ext_vector_type(8)));

#define NB    2
#define RES   64
#define CDIM  32
#define RESO  512
#define SH    66                 // 64 + 1 halo each side
#define SLAB  (3*SH*SH)          // 13068 floats = 51.1 KB LDS

// Mirror reference coordinate->bin math in f32:
// c = linspace(-0.5,0.5,64)[i]; u = c/(1+0.1+10e-4)+0.5; clamp; bin = int(u*512)
__device__ __forceinline__ int coord_bin(int idx) {
    float c = -0.5f + (float)idx * (1.0f / 63.0f);
    float u = c / 1.101f + 0.5f;
    if (u >= 1.0f) u = 1.0f - 1e-5f;
    if (u < 0.0f)  u = 0.0f;
    return (int)(u * (float)RESO);
}

// ---------------------------------------------------------------------------
// Kernel 1: stream 256 MB of zeros with non-temporal stores (bandwidth floor).
// ---------------------------------------------------------------------------
__global__ __launch_bounds__(256) void zero_fill_kernel(float* __restrict__ out,
                                                        long long nvec4) {
    long long i = (long long)blockIdx.x * blockDim.x + threadIdx.x;
    if (i < nvec4) {
        v4f z = {0.0f, 0.0f, 0.0f, 0.0f};
        __builtin_nontemporal_store(z, (v4f*)(out + 4 * i));
    }
}

// ---------------------------------------------------------------------------
// Kernel 2: per (b, plane, d) slice: conv3d (im2col WMMA f32 16x16x4) + bias
// + ReLU, reduce over h, scatter 32ch x 64w cells into the output plane.
// Block = 128 threads = 4 waves; wave w owns w-tile [16w, 16w+15] and both
// 16-channel M tiles.
// ---------------------------------------------------------------------------
__global__ __launch_bounds__(128) void conv_scatter_wmma_kernel(
    const float* __restrict__ x,   const float* __restrict__ col,
    const float* __restrict__ w_p, const float* __restrict__ b_p,
    const float* __restrict__ w_r, const float* __restrict__ b_r,
    const float* __restrict__ w_g, const float* __restrict__ b_g,
    const float* __restrict__ w_b, const float* __restrict__ b_b,
    float* __restrict__ out)
{
    __shared__ float S[SLAB];

    const int wg = blockIdx.x;
    const int d  = wg & 63;          // slice along D
    const int q  = (wg >> 6) & 3;    // output plane: 0=p,1=r,2=b,3=g
    const int b  = wg >> 8;          // batch

    // plane q -> (weights, bias, col channel); output tuple order is (p,r,b,g)
    const float* W  = w_p;
    const float* Bv = b_p;
    int ch = 0;
    if      (q == 1) { W = w_r; Bv = b_r; ch = 0; }
    else if (q == 2) { W = w_b; Bv = b_b; ch = 2; }
    else if (q == 3) { W = w_g; Bv = b_g; ch = 1; }

    // -------- cooperative load of zero-padded 3x66x66 slab into LDS --------
    const int tid = threadIdx.x;
    for (int e = tid; e < SLAB; e += 128) {
        int dz = e / (SH * SH);
        int r  = e - dz * (SH * SH);
        int hh = r / SH;
        int ww = r - hh * SH;
        int dd = d - 1 + dz;
        int h  = hh - 1;
        int w  = ww - 1;
        float v = 0.0f;
        if ((unsigned)dd < 64u && (unsigned)h < 64u && (unsigned)w < 64u) {
            int vi = ((b * 64 + dd) * 64 + h) * 64 + w;
            v = (q == 0) ? x[vi] : col[vi * 3 + ch];
        }
        S[e] = v;
    }
    __syncthreads();

    const int lane = tid & 31;
    const int wt   = tid >> 5;       // w-tile 0..3
    const int w0   = wt * 16;
    const int n    = lane & 15;      // N index / M row within half-wave
    const int hi   = lane >> 4;      // 0: K pair {0,1}, 1: K pair {2,3}

    // -------- A matrices (weights), 7 K-steps x 2 M-tiles, K padded to 28 ----
    // A 16x4 f32 layout: lane L holds row L&15; VGPR0/1 = K {2*hi, 2*hi+1}
    v2f A0[7], A1[7];
    #pragma unroll
    for (int kk = 0; kk < 7; ++kk) {
        int kb = 4 * kk + 2 * hi;
        v2f a0; v2f a1;
        a0[0] = (kb     < 27) ? W[n * 27 + kb]            : 0.0f;
        a0[1] = (kb + 1 < 27) ? W[n * 27 + kb + 1]        : 0.0f;
        a1[0] = (kb     < 27) ? W[(16 + n) * 27 + kb]     : 0.0f;
        a1[1] = (kb + 1 < 27) ? W[(16 + n) * 27 + kb + 1] : 0.0f;
        A0[kk] = a0; A1[kk] = a1;
    }

    // bias per accumulator VGPR: channel = r + 8*hi (C/D layout)
    float bv0[8], bv1[8];
    #pragma unroll
    for (int r = 0; r < 8; ++r) {
        bv0[r] = Bv[r + 8 * hi];
        bv1[r] = Bv[16 + r + 8 * hi];
    }

    // -------- per-lane im2col tap offsets into LDS (constant part) ----------
    // B 4x16 layout: lane (hi,n) VGPR vg = P[2*hi+vg][n]; tap k = 4kk+2hi+vg
    int   cof[7][2];
    float msk[7][2];
    #pragma unroll
    for (int kk = 0; kk < 7; ++kk) {
        #pragma unroll
        for (int vg = 0; vg < 2; ++vg) {
            int k = 4 * kk + 2 * hi + vg;
            if (k < 27) {
                int dz  = k / 9;
                int rem = k - dz * 9;
                int dh  = rem / 3;
                int dw  = rem - dh * 3;
                cof[kk][vg] = dz * SH * SH + dh * SH + (w0 + n + dw);
                msk[kk][vg] = 1.0f;
            } else {
                cof[kk][vg] = 0;      // harmless in-bounds read, masked to 0
                msk[kk][vg] = 0.0f;
            }
        }
    }

    // -------- main loop over h: WMMA GEMM, bias+ReLU, running sum -----------
    v8f acc0 = {}; v8f acc1 = {};
    for (int h = 0; h < 64; ++h) {
        const int hoff = h * SH;
        v2f Bm[7];
        #pragma unroll
        for (int kk = 0; kk < 7; ++kk) {
            v2f bm;
            bm[0] = S[cof[kk][0] + hoff] * msk[kk][0];
            bm[1] = S[cof[kk][1] + hoff] * msk[kk][1];
            Bm[kk] = bm;
        }
        v8f c0 = {}; v8f c1 = {};
        #pragma unroll
        for (int kk = 0; kk < 7; ++kk) {
            c0 = __builtin_amdgcn_wmma_f32_16x16x4_f32(
                     false, A0[kk], false, Bm[kk], (short)0, c0, false, false);
            c1 = __builtin_amdgcn_wmma_f32_16x16x4_f32(
                     false, A1[kk], false, Bm[kk], (short)0, c1, false, false);
        }
        #pragma unroll
        for (int r = 0; r < 8; ++r) {
            acc0[r] += fmaxf(c0[r] + bv0[r], 0.0f);
            acc1[r] += fmaxf(c1[r] + bv1[r], 0.0f);
        }
    }

    // -------- scatter means into the plane ----------------------------------
    const float inv = 1.0f / 64.0f;
    const int   id  = coord_bin(d);        // plane column (from d)
    const int   jw  = coord_bin(w0 + n);   // plane row    (from w, per lane)
    const size_t plane_base =
        (size_t)q * (size_t)(NB * CDIM * RESO * RESO) +
        (size_t)b * (size_t)(CDIM * RESO * RESO);
    #pragma unroll
    for (int r = 0; r < 8; ++r) {
        int c0i = r + 8 * hi;
        int c1i = 16 + r + 8 * hi;
        out[plane_base + ((size_t)c0i * RESO + (size_t)jw) * RESO + (size_t)id] = acc0[r] * inv;
        out[plane_base + ((size_t)c1i * RESO + (size_t)jw) * RESO + (size_t)id] = acc1[r] * inv;
    }
}

extern "C" void kernel_launch(void* const* d_in, const int* in_sizes, int n_in,
                              void* d_out, int out_size, void* d_ws, size_t ws_size,
                              hipStream_t stream) {
    (void)in_sizes; (void)n_in; (void)d_ws; (void)ws_size;
    const float* x   = (const float*)d_in[0];
    const float* col = (const float*)d_in[1];
    const float* w_p = (const float*)d_in[2];
    const float* b_p = (const float*)d_in[3];
    const float* w_r = (const float*)d_in[4];
    const float* b_r = (const float*)d_in[5];
    const float* w_g = (const float*)d_in[6];
    const float* b_g = (const float*)d_in[7];
    const float* w_b = (const float*)d_in[8];
    const float* b_b = (const float*)d_in[9];
    float* out = (float*)d_out;

    // 1) zero the 256 MB output with NT vec4 stores (out_size divisible by 4)
    long long nvec4  = (long long)out_size / 4;
    long long blocks = (nvec4 + 255) / 256;
    zero_fill_kernel<<<(dim3)(unsigned)blocks, 256, 0, stream>>>(out, nvec4);

    // 2) conv + reduce + scatter: one workgroup per (b, plane, d) slice
    conv_scatter_wmma_kernel<<<NB * 4 * RES, 128, 0, stream>>>(
        x, col, w_p, b_p, w_r, b_r, w_g, b_g, w_b, b_b, out);
}